// KoopmanAFTTransitionModel_30305289240685
// MI455X (gfx1250) — compile-verified
//
#include <hip/hip_runtime.h>

// Problem dims (fixed by the reference)
#define NB   4096   // batch B
#define DL   256    // D_LATENT
#define UD   32     // U_DIM
#define AD   64     // AFT_DIM
#define NYD  25     // N_Y
#define NT   64     // T
#define WINN 3

typedef float v2f __attribute__((ext_vector_type(2)));
typedef float v8f __attribute__((ext_vector_type(8)));

// ---- f32 WMMA 16x16x4: D = A(16x4) * B(4x16) + C --------------------------
__device__ __forceinline__ v8f wmma4(v2f a, v2f b, v8f c) {
  return __builtin_amdgcn_wmma_f32_16x16x4_f32(false, a, false, b, (short)0, c,
                                               false, false);
}

// A-frag: out = X @ W^T ; X row-major [*, ldx].
// f32 A 16x4: lanes 0-15 -> M=lane, K=k0+{0,1}; lanes 16-31 -> M=lane-16, K=k0+{2,3}
__device__ __forceinline__ v2f ldA(const float* X, int ldx, int row0, int k0, int lane) {
  int m = lane & 15, ko = (lane >> 4) * 2;
  const float* p = X + (size_t)(row0 + m) * ldx + k0 + ko;
  v2f a; a.x = p[0]; a.y = p[1]; return a;
}
// B-frag: B(k,n) = W[n,k], W row-major [nout, ldw]
__device__ __forceinline__ v2f ldB(const float* W, int ldw, int col0, int k0, int lane) {
  int n = lane & 15, ko = (lane >> 4) * 2;
  const float* p = W + (size_t)(col0 + n) * ldw + k0 + ko;
  v2f b; b.x = p[0]; b.y = p[1]; return b;
}
// Guarded B-frag for nout not multiple of 16 (C/Dm: 25 rows)
__device__ __forceinline__ v2f ldBg(const float* W, int ldw, int col0, int k0,
                                    int lane, int nmax) {
  int n = lane & 15, ko = (lane >> 4) * 2;
  v2f b = {};
  if (col0 + n < nmax) {
    const float* p = W + (size_t)(col0 + n) * ldw + k0 + ko;
    b.x = p[0]; b.y = p[1];
  }
  return b;
}

// order-preserving float<->uint key for atomicMax on signed floats
__device__ __forceinline__ unsigned fkey(float f) {
  unsigned u = __float_as_uint(f);
  return (u & 0x80000000u) ? ~u : (u | 0x80000000u);
}
__device__ __forceinline__ float fkey_inv(unsigned k) {
  unsigned u = (k & 0x80000000u) ? (k & 0x7fffffffu) : ~k;
  return __uint_as_float(u);
}

// zero all per-step max keys once (one key per step -> no per-step reset launch)
__global__ void koop_init_kernel(unsigned* mkeys) { mkeys[threadIdx.x] = 0u; }

// Fused projection step: zlin = z@K^T + (Ut*dt)@L^T ; Q/Kp/Vp = z@{Wq,Wk,Wv}^T
// 7 groups x 256 row-strips; each wave does a 16x64 tile (4 accumulators share
// one A-fragment -> 1.25 b64 loads per WMMA instead of 2).
__global__ __launch_bounds__(256) void koop_proj_kernel(
    const float* __restrict__ z, const float* __restrict__ Ut,
    const float* __restrict__ dtv,
    const float* __restrict__ Km, const float* __restrict__ Lm,
    const float* __restrict__ Wq, const float* __restrict__ Wk,
    const float* __restrict__ Wv,
    float* __restrict__ zlin, float* __restrict__ Qo,
    float* __restrict__ Kp, float* __restrict__ Vp,
    unsigned* __restrict__ mkey) {
  int wave = threadIdx.x >> 5, lane = threadIdx.x & 31;
  int tile = blockIdx.x * 8 + wave;   // 1792 groups total
  int rt = tile / 7, grp = tile % 7;
  int row0 = rt * 16;
  int m = lane & 15, ko = (lane >> 4) * 2;
  int mm = (lane >> 4) * 8, n = lane & 15;

  const float* W;
  int colbase;
  if (grp < 4) { W = Km; colbase = grp * 64; }       // zlin cols [64g, 64g+64)
  else { W = (grp == 4) ? Wq : (grp == 5) ? Wk : Wv; colbase = 0; }

  v8f acc0 = {}, acc1 = {}, acc2 = {}, acc3 = {};
  #pragma unroll 2
  for (int k0 = 0; k0 < DL; k0 += 4) {
    v2f a = ldA(z, DL, row0, k0, lane);              // shared across 4 tiles
    acc0 = wmma4(a, ldB(W, DL, colbase +  0, k0, lane), acc0);
    acc1 = wmma4(a, ldB(W, DL, colbase + 16, k0, lane), acc1);
    acc2 = wmma4(a, ldB(W, DL, colbase + 32, k0, lane), acc2);
    acc3 = wmma4(a, ldB(W, DL, colbase + 48, k0, lane), acc3);
  }

  if (grp < 4) {
    // + (Ut*dt) @ L^T (K=32)
    float ds = dtv[row0 + m];
    #pragma unroll
    for (int k0 = 0; k0 < UD; k0 += 4) {
      const float* p = Ut + (size_t)(row0 + m) * UD + k0 + ko;
      v2f a; a.x = p[0] * ds; a.y = p[1] * ds;
      acc0 = wmma4(a, ldB(Lm, UD, colbase +  0, k0, lane), acc0);
      acc1 = wmma4(a, ldB(Lm, UD, colbase + 16, k0, lane), acc1);
      acc2 = wmma4(a, ldB(Lm, UD, colbase + 32, k0, lane), acc2);
      acc3 = wmma4(a, ldB(Lm, UD, colbase + 48, k0, lane), acc3);
    }
    #pragma unroll
    for (int r = 0; r < 8; ++r) {
      size_t o = (size_t)(row0 + mm + r) * DL + colbase + n;
      zlin[o]      = acc0[r];
      zlin[o + 16] = acc1[r];
      zlin[o + 32] = acc2[r];
      zlin[o + 48] = acc3[r];
    }
  } else {
    float* dst = (grp == 4) ? Qo : (grp == 5) ? Kp : Vp;
    #pragma unroll
    for (int r = 0; r < 8; ++r) {
      size_t o = (size_t)(row0 + mm + r) * AD + n;
      dst[o]      = acc0[r];
      dst[o + 16] = acc1[r];
      dst[o + 32] = acc2[r];
      dst[o + 48] = acc3[r];
    }
    if (grp == 5) {  // per-slot max of K-projection (reference: max over (B, a))
      float mx = fmaxf(fmaxf(acc0[0], acc1[0]), fmaxf(acc2[0], acc3[0]));
      #pragma unroll
      for (int r = 1; r < 8; ++r)
        mx = fmaxf(mx, fmaxf(fmaxf(acc0[r], acc1[r]), fmaxf(acc2[r], acc3[r])));
      #pragma unroll
      for (int off = 16; off > 0; off >>= 1)
        mx = fmaxf(mx, __shfl_xor(mx, off, 32));
      if (lane == 0) atomicMax(mkey, fkey(mx));
    }
  }
}

// Fused: attn (elementwise, LDS) -> z1 = zlin + attn@Wo^T -> Z[t] (+LDS)
//        -> y = z1@C^T + (Ut*dt)@Dm^T -> Y[t].
// One block per 16-row strip; 16 waves (512 threads).
__global__ __launch_bounds__(512) void koop_combine_kernel(
    const float* __restrict__ Qo, const float* __restrict__ Kp,
    const float* __restrict__ Vp, const unsigned* __restrict__ mkeys,
    const float* __restrict__ zlin, const float* __restrict__ Wo,
    const float* __restrict__ Cm, const float* __restrict__ Ut,
    const float* __restrict__ dtv, const float* __restrict__ Dmat,
    float* __restrict__ Zt, float* __restrict__ Yt, int t) {
  __shared__ float attn_s[16 * 68];    // padded: stride 68 -> conflict-free frags
  __shared__ float z1_s[16 * 260];     // padded: stride 260
  int row0 = blockIdx.x * 16;
  int tid = threadIdx.x;

  // ---- phase 1: AFT elementwise combine into LDS ----
  for (int i = tid; i < 16 * AD; i += 512) {
    int r = i >> 6, c = i & 63;
    size_t g = (size_t)(row0 + r) * AD + c;
    float num = 0.f, den = 1e-8f;
    int nv = (t < 2) ? (t + 1) : 3;
    for (int j = 0; j < nv; ++j) {
      int step = t - j, slot = step % WINN;
      float m = fkey_inv(mkeys[step]);  // per-window-slot max, cached at origin step
      float e = __expf(Kp[(size_t)slot * (NB * AD) + g] - m);
      num += e * Vp[(size_t)slot * (NB * AD) + g];
      den += e;
    }
    float q = Qo[g];
    attn_s[r * 68 + c] = (1.f / (1.f + __expf(-q))) * (num / den);
  }
  __syncthreads();

  int wave = tid >> 5, lane = tid & 31;
  int m = lane & 15, ko = (lane >> 4) * 2;
  int mm = (lane >> 4) * 8, n = lane & 15;

  // ---- phase 2: z1 tile per wave (cols [16*wave, 16*wave+16)) ----
  {
    int col0 = wave * 16;
    v8f acc = {};
    #pragma unroll 4
    for (int k0 = 0; k0 < AD; k0 += 4) {
      v2f a; a.x = attn_s[m * 68 + k0 + ko]; a.y = attn_s[m * 68 + k0 + ko + 1];
      acc = wmma4(a, ldB(Wo, AD, col0, k0, lane), acc);
    }
    #pragma unroll
    for (int r = 0; r < 8; ++r) {
      size_t o = (size_t)(row0 + mm + r) * DL + col0 + n;
      float v = acc[r] + zlin[o];
      Zt[o] = v;
      z1_s[(mm + r) * 260 + col0 + n] = v;
    }
  }
  __syncthreads();

  // ---- phase 3: y readout, waves 0-1 (scalar branch keeps EXEC all-ones) ----
  int wv = __builtin_amdgcn_readfirstlane(wave);
  if (wv < 2) {
    int col0 = wv * 16;
    v8f acc = {};
    #pragma unroll 4
    for (int k0 = 0; k0 < DL; k0 += 4) {
      v2f a; a.x = z1_s[m * 260 + k0 + ko]; a.y = z1_s[m * 260 + k0 + ko + 1];
      acc = wmma4(a, ldBg(Cm, DL, col0, k0, lane, NYD), acc);
    }
    float ds = dtv[row0 + m];
    #pragma unroll
    for (int k0 = 0; k0 < UD; k0 += 4) {
      const float* p = Ut + (size_t)(row0 + m) * UD + k0 + ko;
      v2f a; a.x = p[0] * ds; a.y = p[1] * ds;
      acc = wmma4(a, ldBg(Dmat, UD, col0, k0, lane, NYD), acc);
    }
    #pragma unroll
    for (int r = 0; r < 8; ++r) {
      int col = col0 + n;
      if (col < NYD)
        Yt[(size_t)(row0 + mm + r) * NYD + col] = acc[r];
    }
  }
}

extern "C" void kernel_launch(void* const* d_in, const int* in_sizes, int n_in,
                              void* d_out, int out_size, void* d_ws, size_t ws_size,
                              hipStream_t stream) {
  const float* zt = (const float*)d_in[0];
  const float* dt = (const float*)d_in[1];
  const float* U  = (const float*)d_in[2];
  const float* Km = (const float*)d_in[3];
  const float* Lm = (const float*)d_in[4];
  const float* Cm = (const float*)d_in[5];
  const float* Dm = (const float*)d_in[6];
  const float* Wq = (const float*)d_in[7];
  const float* Wk = (const float*)d_in[8];
  const float* Wv = (const float*)d_in[9];
  const float* Wo = (const float*)d_in[10];

  float* Zbase = (float*)d_out;                    // [T, B, DL]
  float* Ybase = Zbase + (size_t)NT * NB * DL;     // [T, B, NYD]

  // workspace carve (~11 MB, L2-resident)
  float* ws   = (float*)d_ws;
  float* zlin = ws;                                // NB*DL
  float* Qo   = zlin + (size_t)NB * DL;            // NB*AD
  float* Kp   = Qo + (size_t)NB * AD;              // 3*NB*AD (circular by t%3)
  float* Vp   = Kp + (size_t)WINN * NB * AD;       // 3*NB*AD
  unsigned* mkeys = (unsigned*)(Vp + (size_t)WINN * NB * AD);  // NT per-step keys

  koop_init_kernel<<<dim3(1), dim3(NT), 0, stream>>>(mkeys);

  for (int t = 0; t < NT; ++t) {
    const float* z  = (t == 0) ? zt : (Zbase + (size_t)(t - 1) * NB * DL);
    const float* Ut = U + (size_t)t * NB * UD;
    int slot = t % WINN;

    koop_proj_kernel<<<dim3(224), dim3(256), 0, stream>>>(
        z, Ut, dt, Km, Lm, Wq, Wk, Wv, zlin, Qo,
        Kp + (size_t)slot * NB * AD, Vp + (size_t)slot * NB * AD, mkeys + t);
    koop_combine_kernel<<<dim3(NB / 16), dim3(512), 0, stream>>>(
        Qo, Kp, Vp, mkeys, zlin, Wo, Cm, Ut, dt, Dm,
        Zbase + (size_t)t * NB * DL, Ybase + (size_t)t * NB * NYD, t);
  }
}